// QuadraticConv2DTranspose_73942156968364
// MI455X (gfx1250) — compile-verified
//
#include <hip/hip_runtime.h>

// ---------------------------------------------------------------------------
// QuadraticConv2DTranspose for MI455X (gfx1250), parity-specialized f32 WMMA.
//
// Parity of (oh,ow) selects 1/2/2/4 active bed-of-nails taps => only
// 3/6/6/15 of 55 features can be nonzero; contracting only those rows is
// exact and cuts FLOPs 6.2x (14.8 -> 2.4 GFLOP, compute-bound).
//
// v4: A-matrix fully register-resident. Each lane computes its pixel's
// feature vector in registers (both lane-halves redundantly, compile-time
// indices) and selects its WMMA A-fragment K-pair directly -- no feature
// LDS, no A ds_loads, no per-cin dscnt chain. Tap values for a whole
// cin-chunk are prefetched as b128 loads (channel-contiguous) before the
// staging barrier. LDS holds only the shared compacted weight slices.
// ---------------------------------------------------------------------------

typedef float v2f __attribute__((ext_vector_type(2)));
typedef float v4f __attribute__((ext_vector_type(4)));
typedef float v8f __attribute__((ext_vector_type(8)));

#define CIN   64
#define COUT  64
#define HIN   32
#define WIN   32
#define HO    64
#define WO    64
#define BATCH 8
#define CB    8      // cin chunk staged in LDS per barrier pair

// Active patch tap t = kh*3+kw per parity class (CLS = PH*2 + PW).
constexpr int TL[4][4] = {{4,0,0,0},{3,5,0,0},{1,7,0,0},{0,2,6,8}};

// Dynamic feature l -> q-index pair, q[NA] = 1 (linear terms). Compile-time.
constexpr int FA[4][16] = {
  {0,0,1,2, 2,2,2,2,2,2,2,2,2,2,2,2},
  {0,0,1,0,1,2,3,3, 3,3,3,3,3,3,3,3},
  {0,0,1,0,1,2,3,3, 3,3,3,3,3,3,3,3},
  {0,0,0,0,1,1,1,2,2,3,0,1,2,3,4,5}};
constexpr int FB[4][16] = {
  {0,1,1,2, 2,2,2,2,2,2,2,2,2,2,2,2},
  {0,1,1,2,2,2,3,3, 3,3,3,3,3,3,3,3},
  {0,1,1,2,2,2,3,3, 3,3,3,3,3,3,3,3},
  {0,1,2,3,1,2,3,2,3,3,4,4,4,4,4,5}};

// Compact feature l -> original weight row l' in [0,55)
// (triu_off=[0,9,17,24,30,35,39,42,44]; quad (i,j): off[i]+(j-i);
//  linear t: 45+t; bias: 54). Rows [0,NV) only; pads are zeroed once.
__device__ __constant__ int WL[4][16] = {
  {30,49,54,54, 54,54,54,54,54,54,54,54,54,54,54,54},
  {24,26,35,48,50,54,54,54, 54,54,54,54,54,54,54,54},
  { 9,15,42,46,52,54,54,54, 54,54,54,54,54,54,54,54},
  { 0, 2, 6, 8,17,21,23,39,41,44,45,47,51,53,54,54}};

template<int CLS>
__device__ __forceinline__ void qct_body(const float* __restrict__ x,
                                         const float* __restrict__ wk,
                                         float* __restrict__ out,
                                         float* s_wt)
{
    constexpr int PH  = CLS >> 1;                            // oh parity
    constexpr int PW  = CLS & 1;                             // ow parity
    constexpr int NA  = (CLS == 0) ? 1 : (CLS == 3) ? 4 : 2; // active taps
    constexpr int LC  = (CLS == 0) ? 4 : (CLS == 3) ? 16 : 8;// padded K/cin
    constexpr int DYN = NA * (NA + 1) / 2 + NA;              // dynamic feats
    constexpr int NV  = DYN + 1;                             // + bias row
    constexpr int PITCH = LC + 2;                            // even
    constexpr int WIT = (NV * COUT + 255) / 256;             // staging iters

    const int tid  = threadIdx.x;     // 0..255
    const int wave = tid >> 5;        // 0..7
    const int lane = tid & 31;
    const int hk   = lane >> 4;       // lane half
    const int l16  = lane & 15;

    // 64 blocks per class: b (8) x y-tiles (4 of 8 rows) x x-tiles (2 of 16)
    const int blk  = blockIdx.x;
    const int b    = blk >> 3;
    const int rt   = (blk >> 1) & 3;
    const int ct   = blk & 1;
    const int wrow = wave >> 1;       // wave sub-tile: 2 y-rows x 8 x-cols
    const int wcol = wave & 1;

    // this lane's feature pixel (M row l16) in class-local (y, xc) coords
    const int y  = rt * 8  + wrow * 2 + (l16 >> 3);   // 0..31
    const int xc = ct * 16 + wcol * 8 + (l16 & 7);    // 0..31
    const int oh = PH + 2 * y;
    const int ow = PW + 2 * xc;

    // --- loop-invariant tap pointers (clamped in-bounds) + 0/1 masks ---
    const float* tp[NA];
    float tm[NA];
    #pragma unroll
    for (int i = 0; i < NA; ++i) {
        const int t  = TL[CLS][i];
        const int r  = oh + t / 3;                 // odd by construction
        const int qc = ow + t % 3;                 // odd by construction
        const int ri = (r >> 1)  > 31 ? 31 : (r >> 1);
        const int qi = (qc >> 1) > 31 ? 31 : (qc >> 1);
        tp[i] = x + ((b * HIN + ri) * WIN + qi) * CIN;
        tm[i] = (r < 64 && qc < 64) ? 1.0f : 0.0f;
    }

    // --- loop-invariant weight-staging addresses ---
    int  wgoff[WIT], wsoff[WIT];
    bool wact[WIT];
    #pragma unroll
    for (int it = 0; it < WIT; ++it) {
        const int idx = tid + it * 256;
        wact[it]  = idx < NV * COUT;
        const int l = idx >> 6;       // < 16, safe for WL
        const int o = idx & 63;
        wgoff[it] = WL[CLS][l] * (CIN * COUT) + o;   // + c*COUT at use
        wsoff[it] = o * PITCH + l;                   // + cc*COUT*PITCH
    }

    // --- one-time: zero the pad weight columns ---
    #pragma unroll
    for (int lp = NV; lp < LC; ++lp)
        for (int idx = tid; idx < CB * COUT; idx += 256)
            s_wt[idx * PITCH + lp] = 0.0f;

    v8f acc0 = {}, acc1 = {}, acc2 = {}, acc3 = {};

    for (int c0 = 0; c0 < CIN; c0 += CB) {
        // --- prefetch this chunk's tap values (channel-contiguous b128s);
        //     overlaps with barrier + weight staging ---
        float qv[NA][CB];
        #pragma unroll
        for (int i = 0; i < NA; ++i) {
            const v4f qa = *(const v4f*)(tp[i] + c0);
            const v4f qb = *(const v4f*)(tp[i] + c0 + 4);
            qv[i][0] = qa.x * tm[i];  qv[i][1] = qa.y * tm[i];
            qv[i][2] = qa.z * tm[i];  qv[i][3] = qa.w * tm[i];
            qv[i][4] = qb.x * tm[i];  qv[i][5] = qb.y * tm[i];
            qv[i][6] = qb.z * tm[i];  qv[i][7] = qb.w * tm[i];
        }

        __syncthreads();   // previous chunk's s_wt reads complete
        // --- stage CB cin slices of compacted weight rows ---
        #pragma unroll
        for (int cc = 0; cc < CB; ++cc) {
            const int cgl = (c0 + cc) * COUT;
            const int csh = cc * COUT * PITCH;
            #pragma unroll
            for (int it = 0; it < WIT; ++it)
                if (wact[it]) s_wt[csh + wsoff[it]] = wk[wgoff[it] + cgl];
        }
        __syncthreads();

        // --- per cin: register-resident features + WMMA (full unroll so all
        //     qv/fr/afrag indices are compile-time -> no scratch) ---
        #pragma unroll
        for (int cc = 0; cc < CB; ++cc) {
            float q[NA + 1];
            #pragma unroll
            for (int i = 0; i < NA; ++i) q[i] = qv[i][cc];
            q[NA] = 1.0f;

            float fr[LC];
            #pragma unroll
            for (int l = 0; l < DYN; ++l)
                fr[l] = q[FA[CLS][l]] * q[FB[CLS][l]];
            fr[DYN] = 1.0f;                       // bias feature
            #pragma unroll
            for (int l = NV; l < LC; ++l) fr[l] = 0.0f;   // pads

            // A-fragment: lane (hk,l16) holds A[l16][k0+2hk], A[l16][k0+2hk+1]
            v2f afrag[LC / 4];
            if (hk == 0) {
                #pragma unroll
                for (int kc = 0; kc < LC / 4; ++kc) {
                    afrag[kc][0] = fr[kc * 4 + 0];
                    afrag[kc][1] = fr[kc * 4 + 1];
                }
            } else {
                #pragma unroll
                for (int kc = 0; kc < LC / 4; ++kc) {
                    afrag[kc][0] = fr[kc * 4 + 2];
                    afrag[kc][1] = fr[kc * 4 + 3];
                }
            }

            const float* wb = &s_wt[cc * COUT * PITCH];
            #pragma unroll
            for (int kc = 0; kc < LC / 4; ++kc) {
                const int k0 = kc * 4 + hk * 2;
                v2f b0 = *(const v2f*)&wb[( 0 + l16) * PITCH + k0];
                v2f b1 = *(const v2f*)&wb[(16 + l16) * PITCH + k0];
                v2f b2 = *(const v2f*)&wb[(32 + l16) * PITCH + k0];
                v2f b3 = *(const v2f*)&wb[(48 + l16) * PITCH + k0];
                acc0 = __builtin_amdgcn_wmma_f32_16x16x4_f32(
                    false, afrag[kc], false, b0, (short)0, acc0, false, false);
                acc1 = __builtin_amdgcn_wmma_f32_16x16x4_f32(
                    false, afrag[kc], false, b1, (short)0, acc1, false, false);
                acc2 = __builtin_amdgcn_wmma_f32_16x16x4_f32(
                    false, afrag[kc], false, b2, (short)0, acc2, false, false);
                acc3 = __builtin_amdgcn_wmma_f32_16x16x4_f32(
                    false, afrag[kc], false, b3, (short)0, acc3, false, false);
            }
        }
    }

    // --- store D: lane holds rows m = v + 8*hk, col = nt*16 + l16 ---
    #pragma unroll
    for (int v = 0; v < 8; ++v) {
        const int m   = v + hk * 8;
        const int ym  = rt * 8  + wrow * 2 + (m >> 3);
        const int xm  = ct * 16 + wcol * 8 + (m & 7);
        const int poh = PH + 2 * ym;
        const int pw_ = PW + 2 * xm;
        const size_t base = (((size_t)(b * HO + poh)) * WO + pw_) * COUT;
        out[base +  0 + l16] = acc0[v];
        out[base + 16 + l16] = acc1[v];
        out[base + 32 + l16] = acc2[v];
        out[base + 48 + l16] = acc3[v];
    }
}

__global__ __launch_bounds__(256)
void QuadraticConv2DTranspose_73942156968364_kernel(
    const float* __restrict__ x,    // [8, 32, 32, 64]
    const float* __restrict__ wk,   // [55, 64, 64]
    float* __restrict__ out)        // [8, 64, 64, 64]
{
    __shared__ float s_wt[CB * COUT * 18];   // max pitch (CLS 3): 36 KB
    switch (blockIdx.y) {
        case 0:  qct_body<0>(x, wk, out, s_wt); break;
        case 1:  qct_body<1>(x, wk, out, s_wt); break;
        case 2:  qct_body<2>(x, wk, out, s_wt); break;
        default: qct_body<3>(x, wk, out, s_wt); break;
    }
}

extern "C" void kernel_launch(void* const* d_in, const int* in_sizes, int n_in,
                              void* d_out, int out_size, void* d_ws, size_t ws_size,
                              hipStream_t stream) {
    (void)in_sizes; (void)n_in; (void)out_size; (void)d_ws; (void)ws_size;
    const float* x  = (const float*)d_in[0];   // [8,32,32,64] f32
    const float* wk = (const float*)d_in[1];   // [1,1,55,64,64] f32
    float* out = (float*)d_out;                // [8,64,64,64] f32

    // 64 blocks per parity class x 4 classes; 256 threads (8 wave32)
    dim3 grid(64, 4);
    dim3 block(256);
    QuadraticConv2DTranspose_73942156968364_kernel<<<grid, block, 0, stream>>>(x, wk, out);
}